// MultiHeadAttention_38706245271716
// MI455X (gfx1250) — compile-verified
//
#include <hip/hip_runtime.h>
#include <hip/hip_bf16.h>

// ---------------------------------------------------------------------------
// Problem constants (reference: B=8, S=1024, H=8, M=64, 7 irrep keys,
// dims {1,1,2,2,2,2,2}; head dim D = 12*8 = 96; output [B,S,12,64]).
// ---------------------------------------------------------------------------
#define Bn 8
#define Sn 1024
#define Hn 8
#define Dn 96          // concat head dim
#define NKEY 7

typedef float v2f __attribute__((ext_vector_type(2)));
typedef float v8f __attribute__((ext_vector_type(8)));
typedef int   vsi4 __attribute__((vector_size(4 * sizeof(int))));

__device__ __forceinline__ v8f wmma_f32_16x16x4(v2f a, v2f b, v8f c) {
    // D = A(16x4, f32) * B(4x16, f32) + C(16x16, f32)
    return __builtin_amdgcn_wmma_f32_16x16x4_f32(false, a, false, b, (short)0, c,
                                                 false, false);
}

__device__ __forceinline__ float rmax16(float v) {
    v = fmaxf(v, __shfl_xor(v, 1));
    v = fmaxf(v, __shfl_xor(v, 2));
    v = fmaxf(v, __shfl_xor(v, 4));
    v = fmaxf(v, __shfl_xor(v, 8));
    return v;
}
__device__ __forceinline__ float rsum16(float v) {
    v += __shfl_xor(v, 1);
    v += __shfl_xor(v, 2);
    v += __shfl_xor(v, 4);
    v += __shfl_xor(v, 8);
    return v;
}

// key -> channel base inside D=96, and irrep dim
__device__ __forceinline__ int key_dim(int key)  { return key < 2 ? 1 : 2; }
__device__ __forceinline__ int key_base(int key) { return key < 2 ? key * 8 : 16 + (key - 2) * 16; }
__device__ __forceinline__ int key_row12(int key){ return key < 2 ? key : 2 + (key - 2) * 2; }

// ---------------------------------------------------------------------------
// CDNA5 async global->LDS copy (tracked by ASYNCcnt), 16B per lane.
// Builtin expects (int32x4 AS1*, int32x4 AS3*, imm offset, imm cpol).
// ---------------------------------------------------------------------------
__device__ __forceinline__ void async_copy_b128(const float* g, float* l) {
#if defined(__has_builtin) && __has_builtin(__builtin_amdgcn_global_load_async_to_lds_b128)
    __builtin_amdgcn_global_load_async_to_lds_b128(
        (__attribute__((address_space(1))) vsi4*)(void*)(float*)g,
        (__attribute__((address_space(3))) vsi4*)(void*)l, 0, 0);
#else
    unsigned laddr = (unsigned)(size_t)(__attribute__((address_space(3))) float*)l;
    asm volatile("global_load_async_to_lds_b128 %0, %1, off"
                 :: "v"(laddr), "v"(g) : "memory");
#endif
}

// ---------------------------------------------------------------------------
// Kernel 1: per-key Q/K/V projections + split-heads + concat into [B,H,S,96].
// Block handles (b, key, 16 seq rows). 256 threads.
// ---------------------------------------------------------------------------
struct QKVArgs {
    const float* x[NKEY];   // x_key: [B,S,d,64]
    const float* Wq;        // [7,64,64]
    const float* Wk;
    const float* Wv;
    float* Q;               // [B,H,S,96]
    float* K;
    float* V;
};

#define WSTR 65             // padded weight row stride (bank-conflict free)
#define TS1 16              // seq rows per block

__global__ __launch_bounds__(256) void qkv_proj_kernel(QKVArgs a) {
    __shared__ float Wl[3 * 64 * WSTR];        // Wq/Wk/Wv of this key
    __shared__ float xsh[TS1 * 2 * 64];        // up to d=2

    const int SCH = Sn / TS1;                  // 64 chunks
    const int b   = blockIdx.x / (NKEY * SCH);
    const int key = (blockIdx.x / SCH) % NKEY;
    const int s0  = (blockIdx.x % SCH) * TS1;
    const int tid = threadIdx.x;

    const int d    = key_dim(key);
    const int base = key_base(key);
    const int dm   = d * 64;

    // load the three 64x64 weight blocks for this key into LDS (padded)
    for (int i = tid; i < 3 * 4096; i += 256) {
        int w = i / 4096;
        int r = i % 4096;
        int o = r >> 6, c = r & 63;
        const float* src = (w == 0 ? a.Wq : (w == 1 ? a.Wk : a.Wv));
        Wl[w * 64 * WSTR + o * WSTR + c] = src[key * 4096 + r];
    }
    // load x tile [TS1, d, 64]
    const float* xg = a.x[key] + ((size_t)b * Sn + s0) * dm;
    for (int i = tid; i < TS1 * dm; i += 256) xsh[i] = xg[i];
    __syncthreads();

    const int tot = TS1 * dm;                  // outputs per projection
    for (int i = tid; i < 3 * tot; i += 256) {
        int proj = i / tot;
        int r    = i % tot;
        int o    = r & 63;
        int rest = r >> 6;
        int dd   = rest % d;
        int sl   = rest / d;

        const float* wrow = &Wl[proj * 64 * WSTR + o * WSTR];
        const float* xrow = &xsh[(sl * d + dd) * 64];
        float acc = 0.f;
#pragma unroll 8
        for (int c = 0; c < 64; ++c) acc += xrow[c] * wrow[c];

        int h = o >> 3, jj = o & 7;
        int ch = base + dd * 8 + jj;
        float* dst = (proj == 0 ? a.Q : (proj == 1 ? a.K : a.V));
        dst[((size_t)(b * Hn + h) * Sn + (s0 + sl)) * Dn + ch] = acc;
    }
}

// ---------------------------------------------------------------------------
// Kernel 2: flash attention per (b,h).  128 threads = 4 waves; each wave owns
// 16 query rows; 64 query rows per block; key tiles of 32, double-buffered
// via async global->LDS copies (ASYNCcnt).
// QO doubles as Q input and O output (each block reads only its own Q rows
// into registers before overwriting them).
// ---------------------------------------------------------------------------
#define KSTR 100           // K tile row stride  (rows 400B -> 16B aligned)
#define VSTR 104           // V tile row stride  (rows 416B; 2*VSTR%64==16)
#define PSTR 36            // P staging row stride (16 x 36 floats per wave)
#define KBUF (32 * KSTR)   // floats per K buffer
#define VBUF (32 * VSTR)   // floats per V buffer

__global__ __launch_bounds__(128) void flash_attn_kernel(float* __restrict__ QO,
                                                         const float* __restrict__ K,
                                                         const float* __restrict__ V) {
    __shared__ float Ks[2 * KBUF];
    __shared__ float Vs[2 * VBUF];
    __shared__ float Pb[4][16 * PSTR];

    const int QBLK = 16;                       // S/64 q-blocks
    const int bh   = blockIdx.x / QBLK;        // 64 (b,h) pairs
    const int qb   = blockIdx.x % QBLK;
    const int tid  = threadIdx.x;
    const int wave = tid >> 5;
    const int lane = tid & 31;
    const int lr   = lane & 15;                // M / N index within subtile
    const int lh   = lane >> 4;                // half-wave -> K-pair select
    const int q0   = qb * 64 + wave * 16;

    const size_t bhBase = (size_t)bh * Sn * Dn;

    // ---- Q fragments (A-layout), pre-scaled by 1/sqrt(96) ----------------
    const float scale = 0.1020620726159658f;
    v2f qf[24];
    {
        const float* qrow = QO + bhBase + (size_t)(q0 + lr) * Dn + 2 * lh;
#pragma unroll
        for (int kk = 0; kk < 24; ++kk) {
            v2f t = *(const v2f*)(qrow + kk * 4);
            qf[kk] = t * scale;
        }
    }

    v8f Oc[6];
#pragma unroll
    for (int i = 0; i < 6; ++i) Oc[i] = (v8f)0.f;
    float mrow[8], lrow[8];
#pragma unroll
    for (int j = 0; j < 8; ++j) { mrow[j] = -3.0e38f; lrow[j] = 0.f; }

    // issue async copies for one 32-key tile (6 K-chunks + 6 V-chunks / thread)
    const float* Kg0 = K + bhBase;
    const float* Vg0 = V + bhBase;
    auto issue_tile = [&](int kt, int buf) {
        const float* Kg = Kg0 + (size_t)kt * 32 * Dn;
        const float* Vg = Vg0 + (size_t)kt * 32 * Dn;
        float* kb = &Ks[buf * KBUF];
        float* vb = &Vs[buf * VBUF];
#pragma unroll
        for (int i = 0; i < 6; ++i) {
            int cg = tid + i * 128;            // 0..767 : 24 x16B chunks per row
            int r  = cg / 24, cc = (cg % 24) * 4;
            async_copy_b128(Kg + r * Dn + cc, kb + r * KSTR + cc);
            async_copy_b128(Vg + r * Dn + cc, vb + r * VSTR + cc);
        }
    };

    const int NT = Sn / 32;
    issue_tile(0, 0);                          // prologue: tile 0 -> buffer 0

    for (int kt = 0; kt < NT; ++kt) {
        const int cur = kt & 1;
        if (kt + 1 < NT) {
            issue_tile(kt + 1, 1 - cur);       // prefetch next tile
            asm volatile("s_wait_asynccnt 0xc" ::: "memory"); // tile kt landed
        } else {
            asm volatile("s_wait_asynccnt 0x0" ::: "memory");
        }
        __syncthreads();                       // all waves' copies visible

        const float* kb0 = &Ks[cur * KBUF];
        const float* vb0 = &Vs[cur * VBUF];

        // ---- S = (Q*scale) K^T : two 16x16 output subtiles ---------------
        v8f Sa[2];
        Sa[0] = (v8f)0.f; Sa[1] = (v8f)0.f;
#pragma unroll
        for (int t = 0; t < 2; ++t) {
            const float* kb = kb0 + (t * 16 + lr) * KSTR + 2 * lh;
#pragma unroll
            for (int kk = 0; kk < 24; ++kk) {
                v2f bf = *(const v2f*)(kb + kk * 4);
                Sa[t] = wmma_f32_16x16x4(qf[kk], bf, Sa[t]);
            }
        }

        // ---- online softmax over the 32 new keys -------------------------
        float alpha[8];
#pragma unroll
        for (int j = 0; j < 8; ++j) {
            float mx = fmaxf(Sa[0][j], Sa[1][j]);
            mx = rmax16(mx);
            float mnew = fmaxf(mrow[j], mx);
            float al   = __expf(mrow[j] - mnew);
            float p0 = __expf(Sa[0][j] - mnew);
            float p1 = __expf(Sa[1][j] - mnew);
            Sa[0][j] = p0;
            Sa[1][j] = p1;
            float rs = rsum16(p0 + p1);
            lrow[j] = lrow[j] * al + rs;
            mrow[j] = mnew;
            alpha[j] = al;
        }
#pragma unroll
        for (int c6 = 0; c6 < 6; ++c6)
#pragma unroll
            for (int j = 0; j < 8; ++j) Oc[c6][j] *= alpha[j];

        // ---- stage P through LDS to convert D-layout -> A-layout ---------
        float* pw = &Pb[wave][0];
#pragma unroll
        for (int t = 0; t < 2; ++t)
#pragma unroll
            for (int j = 0; j < 8; ++j)
                pw[(j + 8 * lh) * PSTR + t * 16 + lr] = Sa[t][j];

        asm volatile("s_wait_dscnt 0x0" ::: "memory");  // same-wave LDS RAW

        // ---- O += P V : 8 K-chunks x 6 channel subtiles ------------------
#pragma unroll
        for (int kk = 0; kk < 8; ++kk) {
            v2f af = *(const v2f*)(pw + lr * PSTR + kk * 4 + 2 * lh);
            const float* vk = vb0 + (kk * 4 + 2 * lh) * VSTR + lr;
#pragma unroll
            for (int c6 = 0; c6 < 6; ++c6) {
                v2f bf;
                bf.x = vk[c6 * 16];
                bf.y = vk[c6 * 16 + VSTR];
                Oc[c6] = wmma_f32_16x16x4(af, bf, Oc[c6]);
            }
        }
        __syncthreads();     // buffer `cur` free before it is refilled
    }

    // ---- normalize and store (overwrites this block's own Q rows) --------
    float inv[8];
#pragma unroll
    for (int j = 0; j < 8; ++j) inv[j] = 1.0f / lrow[j];
#pragma unroll
    for (int c6 = 0; c6 < 6; ++c6)
#pragma unroll
        for (int j = 0; j < 8; ++j)
            QO[bhBase + (size_t)(q0 + j + 8 * lh) * Dn + c6 * 16 + lr] = Oc[c6][j] * inv[j];
}

// ---------------------------------------------------------------------------
// Kernel 3: combine heads + per-key output projection -> out [B,S,12,64].
// Block handles (b, 16 seq rows). 256 threads.
// ---------------------------------------------------------------------------
#define CSTR 772           // padded comb row stride (12*64 + 4)

__global__ __launch_bounds__(256) void out_proj_kernel(const float* __restrict__ Oattn,
                                                       const float* __restrict__ Wo,
                                                       float* __restrict__ out) {
    __shared__ float comb[16 * CSTR];
    const int b  = blockIdx.x >> 6;
    const int s0 = (blockIdx.x & 63) << 4;

    // gather: comb[sl][row12*64 + h*8 + jj] = Oattn[b,h,s0+sl, ch]
    for (int i = threadIdx.x; i < Hn * 16 * Dn; i += 256) {
        int h   = i / (16 * Dn);
        int rem = i % (16 * Dn);
        int sl  = rem / Dn;
        int ch  = rem % Dn;                       // coalesced along ch
        float v = Oattn[((size_t)(b * Hn + h) * Sn + (s0 + sl)) * Dn + ch];
        int key, base;
        if (ch < 16) { key = ch >> 3;              base = key * 8; }
        else         { key = 2 + ((ch - 16) >> 4); base = 16 + (key - 2) * 16; }
        int within = ch - base;
        int dd = within >> 3, jj = within & 7;
        int row12 = key_row12(key) + dd;
        comb[sl * CSTR + row12 * 64 + h * 8 + jj] = v;
    }
    __syncthreads();

    for (int i = threadIdx.x; i < 16 * 12 * 64; i += 256) {
        int sl    = i / 768;
        int r     = i % 768;
        int row12 = r >> 6;
        int oc    = r & 63;
        int key   = row12 < 2 ? row12 : 2 + ((row12 - 2) >> 1);
        const float* w  = Wo + ((size_t)key * 64 + oc) * 64;
        const float* cb = &comb[sl * CSTR + row12 * 64];
        float acc = 0.f;
#pragma unroll 8
        for (int ic = 0; ic < 64; ++ic) acc += cb[ic] * w[ic];
        out[(((size_t)b * Sn + (s0 + sl)) * 12 + row12) * 64 + oc] = acc;
    }
}

// ---------------------------------------------------------------------------
// Launch
// ---------------------------------------------------------------------------
extern "C" void kernel_launch(void* const* d_in, const int* in_sizes, int n_in,
                              void* d_out, int out_size, void* d_ws, size_t ws_size,
                              hipStream_t stream) {
    (void)in_sizes; (void)n_in; (void)out_size; (void)ws_size;

    const size_t PER = (size_t)Bn * Hn * Sn * Dn;   // 6,291,456 floats
    float* QO = (float*)d_ws;                       // Q, later overwritten by O
    float* Kw = QO + PER;
    float* Vw = QO + 2 * PER;

    QKVArgs a;
    for (int i = 0; i < NKEY; ++i) a.x[i] = (const float*)d_in[i];
    a.Wq = (const float*)d_in[7];
    a.Wk = (const float*)d_in[8];
    a.Wv = (const float*)d_in[9];
    a.Q = QO; a.K = Kw; a.V = Vw;

    qkv_proj_kernel<<<dim3(Bn * NKEY * (Sn / TS1)), 256, 0, stream>>>(a);
    flash_attn_kernel<<<dim3(Bn * Hn * (Sn / 64)), 128, 0, stream>>>(QO, Kw, Vw);
    out_proj_kernel<<<dim3(Bn * (Sn / 16)), 256, 0, stream>>>(
        QO, (const float*)d_in[10], (float*)d_out);
}